// DistanceAwareMasking_33663953666648
// MI455X (gfx1250) — compile-verified
//
#include <hip/hip_runtime.h>
#include <hip/hip_bf16.h>

// MI455X / gfx1250. Store-bandwidth-bound kernel (512 MiB fp32 output,
// ~1.4 GFLOP compute): design centers on coalesced b128 NT stores; the tiny
// 4->3->8 MLP is mapped onto V_WMMA_F32_16X16X4_F32 (16 pairs per wave-step).

typedef float v2f __attribute__((ext_vector_type(2)));
typedef float v4f __attribute__((ext_vector_type(4)));
typedef float v8f __attribute__((ext_vector_type(8)));

#define NNODES 4096
#define WAVES_PER_BLOCK 8
#define TILES_PER_WAVE 4          // 16 j's per tile -> 64 j's per wave
#define J_PER_BLOCK (WAVES_PER_BLOCK * TILES_PER_WAVE * 16)   // 512

__global__ __launch_bounds__(256)
void DistanceAwareMasking_kernel(const float* __restrict__ coords,
                                 const float* __restrict__ W1,   // [3,4]
                                 const float* __restrict__ B1v,  // [3]
                                 const float* __restrict__ W2,   // [8,3]
                                 const float* __restrict__ B2v,  // [8]
                                 float* __restrict__ out)        // [N,N,8]
{
    // one 16x8 fp32 transpose tile per wave
    __shared__ float tile[WAVES_PER_BLOCK][16 * 8];

    const int lane = threadIdx.x & 31;
    const int wv   = threadIdx.x >> 5;
    const int i    = blockIdx.y;
    const int m    = lane & 15;          // M-row (A layout) / pair column (B layout)
    const bool lo  = lane < 16;          // lanes 0-15 hold K=0,1 ; 16-31 hold K=2,3
    const int k0   = lo ? 0 : 2;

    // ---------- constant operands, built once ----------
    // A1 = W1 zero-padded to 16x4 (M = hidden unit, K = feature).
    // 32-bit A 16x4 layout: lane m holds (VGPR0: K=k0, VGPR1: K=k0+1).
    v2f a1;
    a1.x = (m < 3) ? W1[m * 4 + k0]     : 0.0f;
    a1.y = (m < 3) ? W1[m * 4 + k0 + 1] : 0.0f;

    // A2 = W2 zero-padded to 16x4 (M = out channel, K = hidden; K=3 pad row = 0).
    v2f a2;
    a2.x = (m < 8 && k0 < 3) ? W2[m * 3 + k0] : 0.0f;   // K = 0 or 2
    a2.y = (m < 8 && lo)     ? W2[m * 3 + 1]  : 0.0f;   // K = 1 valid, K = 3 pad

    // C matrices carry the biases broadcast along columns (rows = M).
    // 16x16 f32 C/D layout: VGPR r holds M=r (lanes 0-15) and M=r+8 (lanes 16-31).
    v8f c1 = {};
    if (lo) { c1.s0 = B1v[0]; c1.s1 = B1v[1]; c1.s2 = B1v[2]; }
    v8f c2 = {};
    if (lo) { c2.s0 = B2v[0]; c2.s1 = B2v[1]; c2.s2 = B2v[2]; c2.s3 = B2v[3];
              c2.s4 = B2v[4]; c2.s5 = B2v[5]; c2.s6 = B2v[6]; c2.s7 = B2v[7]; }

    const float cix = coords[i * 3 + 0];
    const float ciy = coords[i * 3 + 1];
    const float ciz = coords[i * 3 + 2];

    int jbase = blockIdx.x * J_PER_BLOCK + wv * (TILES_PER_WAVE * 16);

    for (int t = 0; t < TILES_PER_WAVE; ++t, jbase += 16) {
        // ---------- features for 16 pairs: B1 = features^T (4x16) ----------
        const int j = jbase + m;
        const float dx = cix - coords[j * 3 + 0];
        const float dy = ciy - coords[j * 3 + 1];
        const float dz = ciz - coords[j * 3 + 2];
        // native v_sqrt_f32 (~1 ulp): avoids the IEEE refinement expansion,
        // irrelevant to accuracy here and cuts per-tile VALU work in half.
        const float dist = __builtin_amdgcn_sqrtf(dx * dx + dy * dy + dz * dz);
        v2f bf;                       // lanes 0-15: rows K=0,1 ; lanes 16-31: K=2,3
        bf.x = lo ? dx : dz;
        bf.y = lo ? dy : dist;

        // ---------- layer 1: D1[hidden][pair] = A1 x B1 + C1 ----------
        v8f d1 = __builtin_amdgcn_wmma_f32_16x16x4_f32(
            false, a1, false, bf, (short)0, c1, false, false);

        // ReLU + repack h^T (4x16) as B2. h rows 0..2 live in d1.s0..s2 at
        // lanes 0-15; row 2 must move to lanes 16-31 (K=2 slot) -> one shuffle.
        // Values are finite (no NaN inputs): plain select -> single max/cndmask.
        const float h0 = d1.s0 > 0.0f ? d1.s0 : 0.0f;
        const float h1 = d1.s1 > 0.0f ? d1.s1 : 0.0f;
        const float h2 = d1.s2 > 0.0f ? d1.s2 : 0.0f;
        const float h2s = __shfl(h2, m, 32);   // lanes 16-31 read lanes 0-15
        v2f bh;
        bh.x = lo ? h0 : h2s;                  // K=0 | K=2
        bh.y = lo ? h1 : 0.0f;                 // K=1 | K=3 (pad)

        // ---------- layer 2: D2[channel][pair] = A2 x B2 + C2 ----------
        v8f d2 = __builtin_amdgcn_wmma_f32_16x16x4_f32(
            false, a2, false, bh, (short)0, c2, false, false);

        // ---------- transpose via LDS, then one contiguous 512B store ----------
        // Valid channels are M=0..7 -> components s0..s7 at lanes 0-15 (pair = m).
        if (lo) {
            v4f q0 = { d2.s0, d2.s1, d2.s2, d2.s3 };
            v4f q1 = { d2.s4, d2.s5, d2.s6, d2.s7 };
            *(v4f*)&tile[wv][m * 8 + 0] = q0;   // ds_store_b128
            *(v4f*)&tile[wv][m * 8 + 4] = q1;   // ds_store_b128
        }
        // Same-wave DS ops are in-order; compiler inserts s_wait_dscnt before use.
        v4f r = *(v4f*)&tile[wv][lane * 4];      // ds_load_b128

        const size_t off = ((size_t)i * NNODES + (size_t)jbase) * 8 + (size_t)lane * 4;
        __builtin_nontemporal_store(r, (v4f*)&out[off]);   // streaming 512 MiB: NT
    }
}

extern "C" void kernel_launch(void* const* d_in, const int* in_sizes, int n_in,
                              void* d_out, int out_size, void* d_ws, size_t ws_size,
                              hipStream_t stream) {
    const float* coords = (const float*)d_in[0];   // [4096,3]
    const float* W1     = (const float*)d_in[1];   // [3,4]
    const float* b1     = (const float*)d_in[2];   // [3]
    const float* W2     = (const float*)d_in[3];   // [8,3]
    const float* b2     = (const float*)d_in[4];   // [8]
    float* out          = (float*)d_out;           // [4096,4096,8]

    dim3 grid(NNODES / J_PER_BLOCK, NNODES);       // (8, 4096)
    DistanceAwareMasking_kernel<<<grid, 256, 0, stream>>>(coords, W1, b1, W2, b2, out);
}